// LossWassersteinDistance_22840636080247
// MI455X (gfx1250) — compile-verified
//
#include <hip/hip_runtime.h>

// LossWassersteinDistance: B=8, C=5, H=W=96, N=9216.
// Dominant cost: column-sum of the 9216x9216 f32 cost matrix (340 MB).
// HBM roofline @23.3 TB/s  => ~14.6 us. Everything else is noise.
// Strategy:
//   K1: colsum partials via V_WMMA_F32_16X16X4_F32 with A=ones (ones^T * cost),
//       non-temporal loads (matrix > L2, streamed once).
//   K2: deterministic reduce of 8 row-group partials -> colsum[9216].
//   K3: per-(b,c) sumexp / onehot count (40 blocks, LDS tree reduce).
//   K4: per-(b,c) wd = sum_j colsum[j]*|t-p| (40 blocks).
//   K5: mean over 40 -> d_out[0].
// No atomics anywhere -> bitwise deterministic across graph replays.

#define NPIX 9216
#define NBATCH 8
#define NCLS 5
#define NBC (NBATCH * NCLS)
#define EPSV 1e-15f
#define ROWGRP 8            // row groups for colsum partials
#define ROWS_PER_GRP (NPIX / ROWGRP)   // 1152

typedef __attribute__((ext_vector_type(2))) float v2f;
typedef __attribute__((ext_vector_type(8))) float v8f;

// ---------------- K1: column-sum partials via WMMA (D = ones(16x4) * B + C) --
// Each wave owns 16 consecutive columns; block = 8 waves = 128 columns.
// grid.x = 9216/128 = 72 column tiles, grid.y = 8 row groups.
// B tile per WMMA = 4 rows x 16 cols. Since A==1 everywhere, any K->lane
// permutation of the B layout yields the identical column sum, so the load
// pattern below is layout-robust. Accumulator VGPR0 lanes 0..15 hold
// D[M=0, N=lane] = finished sums.
__global__ __launch_bounds__(256) void colsum_wmma_kernel(
    const float* __restrict__ cost, float* __restrict__ partial) {
  const int wave = threadIdx.x >> 5;
  const int lane = threadIdx.x & 31;
  const int col  = blockIdx.x * 128 + wave * 16 + (lane & 15);
  const int rsel = lane >> 4;                 // which of two rows this half-wave feeds
  const int row0 = blockIdx.y * ROWS_PER_GRP;

  v2f a;  a.x = 1.0f; a.y = 1.0f;             // A = ones(16x4): all lanes/VGPRs = 1
  v8f c = {};                                  // C/D accumulator (16x16 f32)

  const float* basep = cost + col;
  for (int r = row0; r < row0 + ROWS_PER_GRP; r += 16) {
#pragma unroll
    for (int k = 0; k < 4; ++k) {
      const size_t rr = (size_t)(r + k * 4 + rsel);
      v2f b;
      // streamed once, 340MB > 192MB L2 -> non-temporal
      b.x = __builtin_nontemporal_load(basep + rr * (size_t)NPIX);
      b.y = __builtin_nontemporal_load(basep + (rr + 2) * (size_t)NPIX);
      c = __builtin_amdgcn_wmma_f32_16x16x4_f32(
          /*neg_a=*/false, a, /*neg_b=*/false, b,
          /*c_mod=*/(short)0, c, /*reuse_a=*/false, /*reuse_b=*/false);
    }
  }
  // D rows are all identical; take M=0 (VGPR0, lanes 0..15).
  if (lane < 16) partial[(size_t)blockIdx.y * NPIX + col] = c[0];
}

// ---------------- K2: colsum[j] = sum over 8 row-group partials -------------
__global__ __launch_bounds__(256) void colsum_reduce_kernel(
    const float* __restrict__ partial, float* __restrict__ colsum) {
  const int j = blockIdx.x * 256 + threadIdx.x;
  if (j < NPIX) {
    float s = 0.0f;
#pragma unroll
    for (int r = 0; r < ROWGRP; ++r) s += partial[(size_t)r * NPIX + j];
    colsum[j] = s;
  }
}

// ---------------- K3: per-(b,c) sumexp and onehot count ---------------------
__global__ __launch_bounds__(256) void stats_kernel(
    const float* __restrict__ outputs, const int* __restrict__ targets,
    float* __restrict__ sumexp, float* __restrict__ cnt) {
  const int bc = blockIdx.x;                 // 0..39
  const int b  = bc / NCLS, cl = bc % NCLS;
  const float* o = outputs + (size_t)bc * NPIX;
  const int*   t = targets + (size_t)b * NPIX;
  float se = 0.0f, ct = 0.0f;
  for (int j = threadIdx.x; j < NPIX; j += 256) {
    se += expf(o[j]);
    ct += (t[j] == cl) ? 1.0f : 0.0f;
  }
  __shared__ float s1[256];
  __shared__ float s2[256];
  s1[threadIdx.x] = se;
  s2[threadIdx.x] = ct;
  __syncthreads();
  for (int s = 128; s > 0; s >>= 1) {
    if (threadIdx.x < s) {
      s1[threadIdx.x] += s1[threadIdx.x + s];
      s2[threadIdx.x] += s2[threadIdx.x + s];
    }
    __syncthreads();
  }
  if (threadIdx.x == 0) { sumexp[bc] = s1[0]; cnt[bc] = s2[0]; }
}

// ---------------- K4: wd[b,c] = sum_j colsum[j] * |t - p| -------------------
__global__ __launch_bounds__(256) void wd_kernel(
    const float* __restrict__ outputs, const int* __restrict__ targets,
    const float* __restrict__ colsum, const float* __restrict__ sumexp,
    const float* __restrict__ cnt, float* __restrict__ wd) {
  const int bc = blockIdx.x;
  const int b  = bc / NCLS, cl = bc % NCLS;
  const float* o  = outputs + (size_t)bc * NPIX;
  const int*   tg = targets + (size_t)b * NPIX;
  const float inv_t = 1.0f / (cnt[bc] + EPSV);
  const float inv_p = 1.0f / (sumexp[bc] + EPSV);
  float acc = 0.0f;
  for (int j = threadIdx.x; j < NPIX; j += 256) {
    const float tv = (tg[j] == cl) ? inv_t : 0.0f;
    const float pv = expf(o[j]) * inv_p;
    acc += colsum[j] * fabsf(tv - pv);
  }
  __shared__ float s1[256];
  s1[threadIdx.x] = acc;
  __syncthreads();
  for (int s = 128; s > 0; s >>= 1) {
    if (threadIdx.x < s) s1[threadIdx.x] += s1[threadIdx.x + s];
    __syncthreads();
  }
  if (threadIdx.x == 0) wd[bc] = s1[0];
}

// ---------------- K5: mean over the 40 wd values ----------------------------
__global__ __launch_bounds__(64) void final_kernel(
    const float* __restrict__ wd, float* __restrict__ out) {
  __shared__ float s[64];
  s[threadIdx.x] = (threadIdx.x < NBC) ? wd[threadIdx.x] : 0.0f;
  __syncthreads();
  for (int st = 32; st > 0; st >>= 1) {
    if (threadIdx.x < st) s[threadIdx.x] += s[threadIdx.x + st];
    __syncthreads();
  }
  if (threadIdx.x == 0) out[0] = s[0] * (1.0f / (float)NBC);
}

extern "C" void kernel_launch(void* const* d_in, const int* in_sizes, int n_in,
                              void* d_out, int out_size, void* d_ws, size_t ws_size,
                              hipStream_t stream) {
  const float* outputs = (const float*)d_in[0];     // (8,5,96,96) f32
  const int*   targets = (const int*)d_in[1];       // (8,96,96) int
  const float* cost    = (const float*)d_in[2];     // (9216,9216) f32
  float* out = (float*)d_out;

  float* ws      = (float*)d_ws;
  float* partial = ws;                               // 8*9216 floats
  float* colsum  = partial + (size_t)ROWGRP * NPIX;  // 9216
  float* sumexp  = colsum + NPIX;                    // 40
  float* cnt     = sumexp + NBC;                     // 40
  float* wd      = cnt + NBC;                        // 40
  // total ws use: ~332 KB

  colsum_wmma_kernel<<<dim3(NPIX / 128, ROWGRP), 256, 0, stream>>>(cost, partial);
  colsum_reduce_kernel<<<(NPIX + 255) / 256, 256, 0, stream>>>(partial, colsum);
  stats_kernel<<<NBC, 256, 0, stream>>>(outputs, targets, sumexp, cnt);
  wd_kernel<<<NBC, 256, 0, stream>>>(outputs, targets, colsum, sumexp, cnt, wd);
  final_kernel<<<1, 64, 0, stream>>>(wd, out);
}